// Mamba_attention_18674517803042
// MI455X (gfx1250) — compile-verified
//
#include <hip/hip_runtime.h>

// ===========================================================================
// Mamba + LightMultiHeadAttention + MLP forward for MI455X (gfx1250, wave32).
// GEMMs use V_WMMA_F32_16X16X4_F32 (fp32 WMMA: numerics match fp32 reference
// while engaging the matrix pipes). Fast path: 16x32 tile per wave (one A
// fragment feeds two WMMAs). Guarded path only for the N=40 x_proj GEMM.
// Selective scan: one lane per (b, d_inner, d_state), shfl_xor reduction.
// ===========================================================================

typedef __attribute__((ext_vector_type(2))) float v2f;
typedef __attribute__((ext_vector_type(8))) float v8f;

#define DEV __device__ __forceinline__

DEV float sigmoidf_(float x) { return 1.f / (1.f + __expf(-x)); }
DEV float siluf_(float x)    { return x * sigmoidf_(x); }
DEV float geluf_(float x)    { return 0.5f * x * (1.f + erff(x * 0.7071067811865476f)); }
DEV float softplusf_(float x){ return (x > 20.f) ? x : log1pf(__expf(x)); }

// ---------------------------------------------------------------------------
// Fast GEMM (N % 32 == 0): C[M,N] = A[M,K](lda) * W[N,K]^T(ldw) + bias[N].
// One wave per 16x32 output tile: one A fragment is reused by two WMMAs
// (two v8f accumulators), halving A traffic vs. the 16x16 version.
// Fragment layouts per the CDNA5 ISA:
//   A: lane(m=lane&15, hi=lane>>4) holds A[m, k0+2*hi+{0,1}]        (v2f)
//   B: lane(n=m, hi)               holds W[n, k0+2*hi+{0,1}]        (v2f)
//   D: vgpr i holds C[row0 + i + 8*hi, col + n]                     (v8f)
// ---------------------------------------------------------------------------
__global__ __launch_bounds__(256)
void gemm_f32_wmma2(const float* __restrict__ A, int lda,
                    const float* __restrict__ W, int ldw,
                    const float* __restrict__ bias,
                    float* __restrict__ C, int ldc,
                    int M, int N, int K)
{
    const int lane   = threadIdx.x & 31;
    const int wave   = threadIdx.x >> 5;
    const int tilesN = N >> 5;                       // 32-wide column tiles
    const int tile   = blockIdx.x * (blockDim.x >> 5) + wave;
    if (tile >= (M >> 4) * tilesN) return;           // wave-uniform exit

    const int tm = tile / tilesN, tn = tile - tm * tilesN;
    const int row0 = tm << 4, col0 = tn << 5;
    const int m  = lane & 15;
    const int hi = lane >> 4;

    const float* arow  = A + (size_t)(row0 + m) * lda + 2 * hi;
    const float* wrow0 = W + (size_t)(col0 + m) * ldw + 2 * hi;
    const float* wrow1 = wrow0 + (size_t)16 * ldw;

    v8f acc0 = {0.f, 0.f, 0.f, 0.f, 0.f, 0.f, 0.f, 0.f};
    v8f acc1 = {0.f, 0.f, 0.f, 0.f, 0.f, 0.f, 0.f, 0.f};
    for (int k0 = 0; k0 < K; k0 += 4) {
        v2f a  = *(const v2f*)(arow  + k0);
        v2f b0 = *(const v2f*)(wrow0 + k0);
        v2f b1 = *(const v2f*)(wrow1 + k0);
        acc0 = __builtin_amdgcn_wmma_f32_16x16x4_f32(
                   false, a, false, b0, (short)0, acc0, false, false);
        acc1 = __builtin_amdgcn_wmma_f32_16x16x4_f32(
                   false, a, false, b1, (short)0, acc1, false, false);
    }
    const float bv0 = bias ? bias[col0 + m]      : 0.f;
    const float bv1 = bias ? bias[col0 + 16 + m] : 0.f;
#pragma unroll
    for (int i = 0; i < 8; ++i) {
        float* crow = C + (size_t)(row0 + i + 8 * hi) * ldc;
        crow[col0 + m]      = acc0[i] + bv0;
        crow[col0 + 16 + m] = acc1[i] + bv1;
    }
}

// ---------------------------------------------------------------------------
// Guarded GEMM (arbitrary N, used only for x_proj N=40): 16x16 tile/wave,
// padded-N lanes masked via a 0/1 multiply on the B fragment (no branch, so
// EXEC stays all-ones around the WMMA).
// ---------------------------------------------------------------------------
__global__ __launch_bounds__(256)
void gemm_f32_wmma_g(const float* __restrict__ A, int lda,
                     const float* __restrict__ W, int ldw,
                     const float* __restrict__ bias,
                     float* __restrict__ C, int ldc,
                     int M, int N, int K)
{
    const int lane   = threadIdx.x & 31;
    const int wave   = threadIdx.x >> 5;
    const int tilesN = (N + 15) >> 4;
    const int tile   = blockIdx.x * (blockDim.x >> 5) + wave;
    if (tile >= (M >> 4) * tilesN) return;

    const int tm = tile / tilesN, tn = tile - tm * tilesN;
    const int row0 = tm << 4, col0 = tn << 4;
    const int m  = lane & 15;
    const int hi = lane >> 4;
    const int colW = col0 + m;
    const float wmask = (colW < N) ? 1.f : 0.f;

    const float* arow = A + (size_t)(row0 + m) * lda + 2 * hi;
    const float* wrow = W + (size_t)(colW < N ? colW : 0) * ldw + 2 * hi;

    v8f acc = {0.f, 0.f, 0.f, 0.f, 0.f, 0.f, 0.f, 0.f};
    for (int k0 = 0; k0 < K; k0 += 4) {
        v2f a = *(const v2f*)(arow + k0);
        v2f b = *(const v2f*)(wrow + k0);
        b.x *= wmask; b.y *= wmask;
        acc = __builtin_amdgcn_wmma_f32_16x16x4_f32(
                  false, a, false, b, (short)0, acc, false, false);
    }
    if (colW < N) {
        float bv = bias ? bias[colW] : 0.f;
#pragma unroll
        for (int i = 0; i < 8; ++i)
            C[(size_t)(row0 + i + 8 * hi) * ldc + colW] = acc[i] + bv;
    }
}

// ---------------------------------------------------------------------------
// Causal depthwise conv1d (k=4, left pad 3) + bias + SiLU.
// Input: u = xz[:, :256] (row stride 512). Output uc[B,L,256] compact.
// ---------------------------------------------------------------------------
__global__ __launch_bounds__(256)
void conv_silu_k(const float* __restrict__ xz, const float* __restrict__ w,
                 const float* __restrict__ cb, float* __restrict__ uc)
{
    int idx = blockIdx.x * blockDim.x + threadIdx.x;   // B*L*256 = 2^22
    int d  = idx & 255;
    int l  = (idx >> 8) & 1023;
    int b  = idx >> 18;
    const float* xcol = xz + (size_t)(b << 10) * 512 + d;
    float acc = cb[d];
#pragma unroll
    for (int j = 0; j < 4; ++j) {
        int ls = l - 3 + j;
        if (ls >= 0) acc += xcol[(size_t)ls * 512] * w[d * 4 + j];
    }
    uc[idx] = siluf_(acc);
}

// ---------------------------------------------------------------------------
// Selective scan. One lane per (b, d, n): grid 256 blocks = 16 b * 16 chunks,
// 256 threads = 16 channels * 16 states. h is a scalar register; y reduced
// over the 16 states with shfl_xor (16-aligned groups inside wave32).
// Fuses softplus(dt_raw), +u*Dp and *silu(z) epilogues.
// ---------------------------------------------------------------------------
__global__ __launch_bounds__(256)
void scan_k(const float* __restrict__ dtraw, const float* __restrict__ uc,
            const float* __restrict__ xdb,   const float* __restrict__ xz,
            const float* __restrict__ A_log, const float* __restrict__ Dp,
            float* __restrict__ y)
{
    int t = threadIdx.x;
    int n = t & 15, dloc = t >> 4;
    int b = blockIdx.x >> 4, chunk = blockIdx.x & 15;
    int d = (chunk << 4) + dloc;

    float An  = -__expf(A_log[d * 16 + n]);
    float Dpd = Dp[d];
    float h = 0.f;
    size_t row = (size_t)b << 10;
    for (int l = 0; l < 1024; ++l, ++row) {
        float dtv = softplusf_(dtraw[row * 256 + d]);   // bias added by GEMM
        float uv  = uc[row * 256 + d];
        float Bn  = xdb[row * 40 + 8 + n];
        float Cn  = xdb[row * 40 + 24 + n];
        float dA  = __expf(dtv * An);
        h = dA * h + (dtv * uv) * Bn;
        float yv = h * Cn;
        yv += __shfl_xor(yv, 1, 32);
        yv += __shfl_xor(yv, 2, 32);
        yv += __shfl_xor(yv, 4, 32);
        yv += __shfl_xor(yv, 8, 32);
        if (n == 0) {
            float zv = xz[row * 512 + 256 + d];
            y[row * 256 + d] = (yv + uv * Dpd) * siluf_(zv);
        }
    }
}

// ---------------------------------------------------------------------------
// LayerNorm over D=128 with fused combine. MODE 0: LN(a+b); 1: LN(gelu(a)*b);
// 2: LN(gelu(a)+gelu(b)). One 128-thread block (4 waves) per row; out stride
// ldo lets e1/e2 write straight into the interleaved `cat` buffer.
// ---------------------------------------------------------------------------
template <int MODE>
__global__ __launch_bounds__(128)
void ln_k(const float* __restrict__ a, const float* __restrict__ b,
          const float* __restrict__ g, const float* __restrict__ be,
          float* __restrict__ out, int ldo)
{
    int row = blockIdx.x, t = threadIdx.x;
    float va = a[(size_t)row * 128 + t];
    float vb = b[(size_t)row * 128 + t];
    float x;
    if (MODE == 0)      x = va + vb;
    else if (MODE == 1) x = geluf_(va) * vb;
    else                x = geluf_(va) + geluf_(vb);

    __shared__ float r1[4], r2[4];
    float s = x;
    for (int o = 16; o > 0; o >>= 1) s += __shfl_down(s, o, 32);
    if ((t & 31) == 0) r1[t >> 5] = s;
    __syncthreads();
    float mean = (r1[0] + r1[1] + r1[2] + r1[3]) * (1.f / 128.f);
    float xc = x - mean;
    float s2 = xc * xc;
    for (int o = 16; o > 0; o >>= 1) s2 += __shfl_down(s2, o, 32);
    if ((t & 31) == 0) r2[t >> 5] = s2;
    __syncthreads();
    float var = (r2[0] + r2[1] + r2[2] + r2[3]) * (1.f / 128.f);
    out[(size_t)row * ldo + t] = xc * rsqrtf(var + 1e-12f) * g[t] + be[t];
}

// Interest-pool logits for both k and v:  l*[b,l,ki] = t . theta[:,ki]
__global__ __launch_bounds__(256)
void pool_logits_k(const float* __restrict__ k, const float* __restrict__ v,
                   const float* __restrict__ thk, const float* __restrict__ thv,
                   float* __restrict__ lk, float* __restrict__ lv)
{
    int idx = blockIdx.x * blockDim.x + threadIdx.x;   // M*8
    int ki = idx & 7;
    size_t row = (size_t)(idx >> 3);
    const float* kr = k + row * 128;
    const float* vr = v + row * 128;
    float sk = 0.f, sv = 0.f;
    for (int d = 0; d < 128; ++d) {
        sk += kr[d] * thk[d * 8 + ki];
        sv += vr[d] * thv[d * 8 + ki];
    }
    lk[idx] = sk; lv[idx] = sv;
}

// In-place softmax over L=1024 for [..., L, 8] tensors. Column c has base
// (c>>3)*8192 + (c&7), stride 8. Block = 256 threads, 4 elements each.
__global__ __launch_bounds__(256)
void softmax_l_k(float* __restrict__ data)
{
    int c = blockIdx.x, t = threadIdx.x;
    size_t base = (size_t)(c >> 3) * 8192 + (c & 7);
    float vals[4];
    float mx = -1e30f;
#pragma unroll
    for (int i = 0; i < 4; ++i) {
        vals[i] = data[base + (size_t)(t + 256 * i) * 8];
        mx = fmaxf(mx, vals[i]);
    }
    __shared__ float sm[8];
    float r = mx;
    for (int o = 16; o > 0; o >>= 1) r = fmaxf(r, __shfl_down(r, o, 32));
    if ((t & 31) == 0) sm[t >> 5] = r;
    __syncthreads();
    float MX = sm[0];
#pragma unroll
    for (int i = 1; i < 8; ++i) MX = fmaxf(MX, sm[i]);
    __syncthreads();
    float ssum = 0.f;
#pragma unroll
    for (int i = 0; i < 4; ++i) { vals[i] = __expf(vals[i] - MX); ssum += vals[i]; }
    for (int o = 16; o > 0; o >>= 1) ssum += __shfl_down(ssum, o, 32);
    if ((t & 31) == 0) sm[t >> 5] = ssum;
    __syncthreads();
    float SUM = sm[0] + sm[1] + sm[2] + sm[3] + sm[4] + sm[5] + sm[6] + sm[7];
    float inv = 1.f / SUM;
#pragma unroll
    for (int i = 0; i < 4; ++i)
        data[base + (size_t)(t + 256 * i) * 8] = vals[i] * inv;
}

// pooled[b,ki,d] = sum_l t[b,l,d] * Dm[b,l,ki].  Block = (b,ki), thread = d.
__global__ __launch_bounds__(128)
void pool_reduce_k(const float* __restrict__ t, const float* __restrict__ Dm,
                   float* __restrict__ out)
{
    int b = blockIdx.x >> 3, ki = blockIdx.x & 7, d = threadIdx.x;
    size_t rb = (size_t)b << 10;
    float s = 0.f;
    for (int l = 0; l < 1024; ++l)
        s += t[(rb + l) * 128 + d] * Dm[(rb + l) * 8 + ki];
    out[(size_t)blockIdx.x * 128 + d] = s;
}

// scores[b,h,l,ki] = (q_head . kp_head) / sqrt(32)
__global__ __launch_bounds__(256)
void scores_k(const float* __restrict__ q, const float* __restrict__ kp,
              float* __restrict__ sc)
{
    int idx = blockIdx.x * blockDim.x + threadIdx.x;   // 2^19
    int ki = idx & 7, l = (idx >> 3) & 1023, h = (idx >> 13) & 3, b = idx >> 15;
    const float* qr = q + ((size_t)(b << 10) + l) * 128 + h * 32;
    const float* kr = kp + (size_t)(b * 8 + ki) * 128 + h * 32;
    float s = 0.f;
#pragma unroll
    for (int j = 0; j < 32; ++j) s += qr[j] * kr[j];
    sc[idx] = s * 0.17677669529663687f;                // 1/sqrt(32)
}

// ctx[b,l,h*32+j] = sum_ki probs[b,h,l,ki] * vp[b,ki,h*32+j]
__global__ __launch_bounds__(256)
void ctx_k(const float* __restrict__ pr, const float* __restrict__ vp,
           float* __restrict__ ctx)
{
    int idx = blockIdx.x * blockDim.x + threadIdx.x;   // 2^21
    int j = idx & 31, h = (idx >> 5) & 3, l = (idx >> 7) & 1023, b = idx >> 17;
    const float* pc = pr + (((size_t)(b * 4 + h) << 10) + l) * 8;
    const float* vc = vp + (size_t)(b * 8) * 128 + h * 32 + j;
    float s = 0.f;
#pragma unroll
    for (int ki = 0; ki < 8; ++ki) s += pc[ki] * vc[(size_t)ki * 128];
    ctx[((size_t)(b << 10) + l) * 128 + h * 32 + j] = s;
}

// ===========================================================================
extern "C" void kernel_launch(void* const* d_in, const int* in_sizes, int n_in,
                              void* d_out, int out_size, void* d_ws, size_t ws_size,
                              hipStream_t stream)
{
    (void)in_sizes; (void)n_in; (void)out_size; (void)ws_size;
    // setup_inputs() dict order: item_emb, then params in insertion order.
    const float* item      = (const float*)d_in[0];
    const float* in_proj_w = (const float*)d_in[1];
    const float* conv_w    = (const float*)d_in[2];
    const float* conv_b    = (const float*)d_in[3];
    const float* x_proj_w  = (const float*)d_in[4];
    const float* dt_proj_w = (const float*)d_in[5];
    const float* dt_proj_b = (const float*)d_in[6];
    const float* A_log     = (const float*)d_in[7];
    const float* Dp        = (const float*)d_in[8];
    const float* out_proj_w= (const float*)d_in[9];
    const float* Wq = (const float*)d_in[10];
    const float* Wk = (const float*)d_in[11];
    const float* Wv = (const float*)d_in[12];
    const float* bq = (const float*)d_in[13];
    const float* bk = (const float*)d_in[14];
    const float* bv = (const float*)d_in[15];
    const float* theta_k = (const float*)d_in[16];
    const float* theta_v = (const float*)d_in[17];
    const float* W1 = (const float*)d_in[18];
    const float* b1 = (const float*)d_in[19];
    const float* W2 = (const float*)d_in[20];
    const float* b2 = (const float*)d_in[21];
    const float* W3 = (const float*)d_in[22];
    const float* b3 = (const float*)d_in[23];
    const float* ln1_g = (const float*)d_in[24];
    const float* ln1_b = (const float*)d_in[25];
    const float* ln2_g = (const float*)d_in[26];
    const float* ln2_b = (const float*)d_in[27];
    const float* ln3_g = (const float*)d_in[28];
    const float* ln3_b = (const float*)d_in[29];
    const float* attn_g = (const float*)d_in[30];
    const float* attn_b = (const float*)d_in[31];

    const int M = 16384;                    // B*L
    float* ws = (float*)d_ws;
    float* e  = ws;                         // M*128, persists across phases
    float* S  = ws + (size_t)M * 128;       // scratch, reused by phase 2
    // phase 1 (Mamba)
    float* xz  = S;                         // M*512
    float* uc  = xz  + (size_t)M * 512;     // M*256
    float* xdb = uc  + (size_t)M * 256;     // M*40
    float* dt  = xdb + (size_t)M * 40;      // M*256 (pre-softplus)
    float* y   = dt  + (size_t)M * 256;     // M*256
    float* mo  = y   + (size_t)M * 256;     // M*128
    // phase 2 (attention + MLP), overlays phase-1 scratch (smaller footprint)
    float* q  = S;
    float* k  = q  + (size_t)M * 128;
    float* v  = k  + (size_t)M * 128;
    float* lk = v  + (size_t)M * 128;       // M*8
    float* lv = lk + (size_t)M * 8;         // M*8
    float* kp = lv + (size_t)M * 8;         // 16*8*128
    float* vp = kp + 16 * 8 * 128;
    float* sc = vp + 16 * 8 * 128;          // 16*4*1024*8
    float* cx = sc + 16 * 4 * 1024 * 8;     // M*128
    float* cat= cx + (size_t)M * 128;       // M*256 ([e1 | e2])
    float* t1 = cat+ (size_t)M * 256;       // M*128
    float* r2 = t1 + (size_t)M * 128;       // M*128
    float* r3 = r2 + (size_t)M * 128;       // M*128

    auto gemm = [&](const float* A, int lda, const float* W, int ldw,
                    const float* bias, float* C, int ldc, int N, int K) {
        if ((N & 31) == 0) {                // fast path: 16x32 tile per wave
            int tiles  = (M / 16) * (N / 32);
            int blocks = (tiles + 7) / 8;   // 8 waves per 256-thread block
            gemm_f32_wmma2<<<blocks, 256, 0, stream>>>(A, lda, W, ldw, bias,
                                                       C, ldc, M, N, K);
        } else {                            // guarded path (x_proj, N=40)
            int tiles  = (M / 16) * ((N + 15) / 16);
            int blocks = (tiles + 7) / 8;
            gemm_f32_wmma_g<<<blocks, 256, 0, stream>>>(A, lda, W, ldw, bias,
                                                        C, ldc, M, N, K);
        }
    };

    // ---- Mamba block ----
    gemm(item, 128, in_proj_w, 128, nullptr, xz, 512, 512, 128);     // xz
    conv_silu_k<<<(M * 256) / 256, 256, 0, stream>>>(xz, conv_w, conv_b, uc);
    gemm(uc, 256, x_proj_w, 256, nullptr, xdb, 40, 40, 256);         // dt|B|C
    gemm(xdb, 40, dt_proj_w, 8, dt_proj_b, dt, 256, 256, 8);         // dt raw
    scan_k<<<256, 256, 0, stream>>>(dt, uc, xdb, xz, A_log, Dp, y);
    gemm(y, 256, out_proj_w, 256, nullptr, mo, 128, 128, 256);
    ln_k<0><<<M, 128, 0, stream>>>(mo, item, ln1_g, ln1_b, e, 128);  // e

    // ---- Light multi-head attention ----
    gemm(e, 128, Wq, 128, bq, q, 128, 128, 128);
    gemm(e, 128, Wk, 128, bk, k, 128, 128, 128);
    gemm(e, 128, Wv, 128, bv, v, 128, 128, 128);
    pool_logits_k<<<(M * 8) / 256, 256, 0, stream>>>(k, v, theta_k, theta_v, lk, lv);
    softmax_l_k<<<128, 256, 0, stream>>>(lk);
    softmax_l_k<<<128, 256, 0, stream>>>(lv);
    pool_reduce_k<<<128, 128, 0, stream>>>(k, lk, kp);
    pool_reduce_k<<<128, 128, 0, stream>>>(v, lv, vp);
    scores_k<<<(16 * 4 * 1024 * 8) / 256, 256, 0, stream>>>(q, kp, sc);
    softmax_l_k<<<512, 256, 0, stream>>>(sc);
    ctx_k<<<(M * 128) / 256, 256, 0, stream>>>(sc, vp, cx);
    ln_k<0><<<M, 128, 0, stream>>>(cx, e, attn_g, attn_b, cat, 256); // e1 -> cat[:, :128]

    // ---- MLPs ----
    gemm(cat, 256, W1, 128, b1, t1, 128, 128, 128);
    ln_k<1><<<M, 128, 0, stream>>>(t1, e, ln2_g, ln2_b, cat + 128, 256); // e2 -> cat[:,128:]
    gemm(cat, 256, W2, 256, b2, r2, 128, 128, 256);
    gemm(item, 128, W3, 128, b3, r3, 128, 128, 128);
    ln_k<2><<<M, 128, 0, stream>>>(r2, r3, ln3_g, ln3_b, (float*)d_out, 128);
}